// ReverseGraphAttentionLayer_28853590295106
// MI455X (gfx1250) — compile-verified
//
#include <hip/hip_runtime.h>

#define B_ 16
#define N_ 1024
#define H_ 768
#define NEG_INF_F (-9000000000000000.0f)
#define SLOPE_F 0.2f

typedef float v2f __attribute__((ext_vector_type(2)));
typedef float v8f __attribute__((ext_vector_type(8)));

#define A_STRIDE 36    // A tile row pitch (floats): conflict-free b64 frag loads, 16B-aligned rows
#define B_PITCH  264   // B tile k-quad pitch (floats): [k>>2][4*n + (k&3)] interleave + pad

// ---------------------------------------------------------------------------
// Kernel 1: node_vec[b] = adj[b] (1024x1024) @ x[b] (1024x768), f32 WMMA.
// Block = 256 threads (8 waves). Block tile: 128 rows x 64 cols. K-block = 32.
// Double-buffered LDS filled with GLOBAL_LOAD_ASYNC_TO_LDS (ASYNCcnt-tracked),
// so the 32 v_wmma_f32_16x16x4_f32 per K-block overlap the next block's DMA.
// ---------------------------------------------------------------------------
__global__ __launch_bounds__(256) void gemm_adj_x(const float* __restrict__ adj,
                                                  const float* __restrict__ x,
                                                  float* __restrict__ nodevec) {
  __shared__ float ldsA[2][128 * A_STRIDE];
  __shared__ float ldsB[2][8 * B_PITCH];

  const int tid  = threadIdx.x;
  const int lane = tid & 31;
  const int wave = tid >> 5;
  const int b    = blockIdx.z;
  const int row0 = blockIdx.y * 128;
  const int col0 = blockIdx.x * 64;

  const float* __restrict__ Ab = adj + (size_t)b * N_ * N_;
  const float* __restrict__ Bb = x   + (size_t)b * N_ * H_;
  float* __restrict__ Cb       = nodevec + (size_t)b * N_ * H_;

  // 32-bit LDS byte addresses for the async-to-LDS VDST operand
  // (flat->LDS keeps the LDS offset in addr[31:0]).
  const unsigned aBase0 = (unsigned)(unsigned long long)(const void*)&ldsA[0][0];
  const unsigned aBase1 = (unsigned)(unsigned long long)(const void*)&ldsA[1][0];
  const unsigned bBase0 = (unsigned)(unsigned long long)(const void*)&ldsB[0][0];
  const unsigned bBase1 = (unsigned)(unsigned long long)(const void*)&ldsB[1][0];

  v8f acc0 = {}, acc1 = {}, acc2 = {}, acc3 = {};

  const int mrow  = wave * 16 + (lane & 15); // A-frag row within 128-row tile
  const int khalf = (lane >> 4) << 1;        // 0 (lanes 0-15) or 2 (lanes 16-31)
  const int nlane = lane & 15;

  // Stage one 128x32 A tile + 32x64 B tile into LDS buffer `buf` asynchronously.
  auto stage = [&](int kb, int buf) {
    const unsigned ab = buf ? aBase1 : aBase0;
    const unsigned bb = buf ? bBase1 : bBase0;
    // A: 1024 float4 = 4 per thread, contiguous 16B on both sides -> async b128
#pragma unroll
    for (int it = 0; it < 4; ++it) {
      const int idx = tid + it * 256;          // 0..1023
      const int r   = idx >> 3;                // 0..127
      const int seg = (idx & 7) << 2;          // 0..28
      const unsigned long long gp =
          (unsigned long long)(const void*)(Ab + (size_t)(row0 + r) * N_ + kb + seg);
      const unsigned lp = ab + (unsigned)((r * A_STRIDE + seg) * 4);
      asm volatile("global_load_async_to_lds_b128 %0, %1, off"
                   :: "v"(lp), "v"(gp) : "memory");
    }
    // B: 2048 floats = 8 per thread; LDS side k-interleaved -> async b32 scatter
#pragma unroll
    for (int it = 0; it < 8; ++it) {
      const int idx = tid + it * 256;          // 0..2047
      const int k   = idx >> 6;                // 0..31
      const int n   = idx & 63;                // 0..63 (coalesced across lanes)
      const unsigned long long gp =
          (unsigned long long)(const void*)(Bb + (size_t)(kb + k) * H_ + col0 + n);
      const unsigned lp = bb + (unsigned)((((k >> 2) * B_PITCH) + n * 4 + (k & 3)) * 4);
      asm volatile("global_load_async_to_lds_b32 %0, %1, off"
                   :: "v"(lp), "v"(gp) : "memory");
    }
  };

  stage(0, 0);
  asm volatile("s_wait_asynccnt 0" ::: "memory");
  __syncthreads();

  for (int kb = 0; kb < N_; kb += 32) {
    const int cur = (kb >> 5) & 1;
    if (kb + 32 < N_) stage(kb + 32, cur ^ 1);   // prefetch next block via async DMA

    const float* lA = &ldsA[cur][0];
    const float* lB = &ldsB[cur][0];

#pragma unroll
    for (int k2 = 0; k2 < 32; k2 += 4) {
      // A fragment (16x4): v0 = A[m][k2+khalf], v1 = A[m][k2+khalf+1] -> one b64
      const v2f av = *(const v2f*)(lA + mrow * A_STRIDE + k2 + khalf);
      // B fragments: interleaved layout makes {kk,kk+1} contiguous -> b64 each
      const float* bp = lB + (k2 >> 2) * B_PITCH + nlane * 4 + khalf;
      const v2f bv0 = *(const v2f*)(bp + 0);     // n-tile 0 (cols  0..15)
      const v2f bv1 = *(const v2f*)(bp + 64);    // n-tile 1 (cols 16..31)
      const v2f bv2 = *(const v2f*)(bp + 128);   // n-tile 2 (cols 32..47)
      const v2f bv3 = *(const v2f*)(bp + 192);   // n-tile 3 (cols 48..63)

      acc0 = __builtin_amdgcn_wmma_f32_16x16x4_f32(false, av, false, bv0, (short)0, acc0, false, false);
      acc1 = __builtin_amdgcn_wmma_f32_16x16x4_f32(false, av, false, bv1, (short)0, acc1, false, false);
      acc2 = __builtin_amdgcn_wmma_f32_16x16x4_f32(false, av, false, bv2, (short)0, acc2, false, false);
      acc3 = __builtin_amdgcn_wmma_f32_16x16x4_f32(false, av, false, bv3, (short)0, acc3, false, false);
    }

    asm volatile("s_wait_asynccnt 0" ::: "memory");  // own async DMA done
    __syncthreads();                                  // everyone's done -> swap
  }

  // ---- store: C/D layout — VGPR r: lanes 0-15 -> M=r, lanes 16-31 -> M=r+8 ----
  const int rbase = row0 + wave * 16 + ((lane >> 4) << 3);
  const int cbase = col0 + nlane;
#pragma unroll
  for (int r = 0; r < 8; ++r) {
    float* cp = Cb + (size_t)(rbase + r) * H_ + cbase;
    cp[0]  = acc0[r];
    cp[16] = acc1[r];
    cp[32] = acc2[r];
    cp[48] = acc3[r];
  }
}

// ---------------------------------------------------------------------------
// Kernel 2a: wa1[f] = sum_j W[f][j]*a[j], wa2[f] = sum_j W[f][j]*a[H+j]
// ---------------------------------------------------------------------------
__global__ __launch_bounds__(128) void compute_wa(const float* __restrict__ W,
                                                  const float* __restrict__ a,
                                                  float* __restrict__ wa) {
  const int f = blockIdx.x * 128 + threadIdx.x;  // grid = 6 -> 768 threads
  const float* wr = W + (size_t)f * H_;
  float acc1 = 0.f, acc2 = 0.f;
  for (int j = 0; j < H_; ++j) {
    const float w = wr[j];
    acc1 += w * a[j];
    acc2 += w * a[H_ + j];
  }
  wa[f]       = acc1;
  wa[H_ + f]  = acc2;
}

// ---------------------------------------------------------------------------
// Kernel 2b: s1/s2[b,n] = dot(node_vec[b,n,:], wa1/wa2). One wave32 per row.
// ---------------------------------------------------------------------------
__global__ __launch_bounds__(256) void compute_s(const float* __restrict__ nodevec,
                                                 const float* __restrict__ wa,
                                                 float* __restrict__ s1,
                                                 float* __restrict__ s2) {
  const int lane = threadIdx.x & 31;
  const int wave = threadIdx.x >> 5;
  const int row  = blockIdx.x * 8 + wave;        // 0 .. B*N-1
  const float* nv = nodevec + (size_t)row * H_;
  float a1 = 0.f, a2 = 0.f;
  for (int f = lane; f < H_; f += 32) {
    const float v = nv[f];
    a1 += v * wa[f];
    a2 += v * wa[H_ + f];
  }
#pragma unroll
  for (int off = 16; off > 0; off >>= 1) {
    a1 += __shfl_xor(a1, off, 32);
    a2 += __shfl_xor(a2, off, 32);
  }
  if (lane == 0) { s1[row] = a1; s2[row] = a2; }
}

// ---------------------------------------------------------------------------
// Kernel 3: masked leaky-relu scores + row softmax. One block per (b,i) row.
// ---------------------------------------------------------------------------
__global__ __launch_bounds__(256) void attn_softmax(const float* __restrict__ adj,
                                                    const float* __restrict__ nl,
                                                    const float* __restrict__ s1,
                                                    const float* __restrict__ s2,
                                                    float* __restrict__ attn) {
  __shared__ float red[8];
  const int b = blockIdx.x >> 10;     // / N_
  const int i = blockIdx.x & (N_ - 1);
  const int tid = threadIdx.x, lane = tid & 31, wave = tid >> 5;

  const float s1i = s1[b * N_ + i];
  const float nli = nl[b * N_ + i];
  const float* arow = adj + ((size_t)b * N_ + i) * N_;

  float sc[4];
  float m = NEG_INF_F;
#pragma unroll
  for (int t = 0; t < 4; ++t) {
    const int j = tid + t * 256;
    const float av = arow[j];
    float e = s1i + s2[b * N_ + j];
    e = e > 0.f ? e : SLOPE_F * e;
    const bool valid = (av > 1e-6f) && (nli * nl[b * N_ + j] > 1e-6f);
    sc[t] = valid ? e : NEG_INF_F;
    m = fmaxf(m, sc[t]);
  }
#pragma unroll
  for (int off = 16; off > 0; off >>= 1) m = fmaxf(m, __shfl_xor(m, off, 32));
  if (lane == 0) red[wave] = m;
  __syncthreads();
  float bm = red[0];
#pragma unroll
  for (int w = 1; w < 8; ++w) bm = fmaxf(bm, red[w]);

  float s = 0.f, ex[4];
#pragma unroll
  for (int t = 0; t < 4; ++t) { ex[t] = __expf(sc[t] - bm); s += ex[t]; }
#pragma unroll
  for (int off = 16; off > 0; off >>= 1) s += __shfl_xor(s, off, 32);
  __syncthreads();
  if (lane == 0) red[wave] = s;
  __syncthreads();
  float bs = 0.f;
#pragma unroll
  for (int w = 0; w < 8; ++w) bs += red[w];
  const float inv = 1.0f / bs;

  float* orow = attn + ((size_t)b * N_ + i) * N_;
#pragma unroll
  for (int t = 0; t < 4; ++t) orow[tid + t * 256] = ex[t] * inv;
}

// ---------------------------------------------------------------------------
extern "C" void kernel_launch(void* const* d_in, const int* in_sizes, int n_in,
                              void* d_out, int out_size, void* d_ws, size_t ws_size,
                              hipStream_t stream) {
  (void)in_sizes; (void)n_in; (void)out_size; (void)ws_size;

  const float* x   = (const float*)d_in[0];  // (B,N,H)
  const float* nl  = (const float*)d_in[1];  // (B,N)
  const float* adj = (const float*)d_in[2];  // (B,N,N)
  const float* W   = (const float*)d_in[3];  // (H,H)
  const float* a   = (const float*)d_in[4];  // (2H,1)

  float* nodevec = (float*)d_out;                          // B*N*H
  float* attn    = (float*)d_out + (size_t)B_ * N_ * H_;   // B*N*N

  float* wa = (float*)d_ws;          // 2*H
  float* s1 = wa + 2 * H_;           // B*N
  float* s2 = s1 + B_ * N_;          // B*N

  gemm_adj_x  <<<dim3(H_ / 64, N_ / 128, B_), 256, 0, stream>>>(adj, x, nodevec);
  compute_wa  <<<dim3(H_ / 128),              128, 0, stream>>>(W, a, wa);
  compute_s   <<<dim3(B_ * N_ / 8),           256, 0, stream>>>(nodevec, wa, s1, s2);
  attn_softmax<<<dim3(B_ * N_),               256, 0, stream>>>(adj, nl, s1, s2, attn);
}